// SSD_9277129360040
// MI455X (gfx1250) — compile-verified
//
#include <hip/hip_runtime.h>
#include <stdint.h>

typedef float v2f  __attribute__((ext_vector_type(2)));
typedef float v8f  __attribute__((ext_vector_type(8)));
typedef unsigned int u32x4 __attribute__((ext_vector_type(4)));
typedef int   i32x4 __attribute__((ext_vector_type(4)));
typedef int   i32x8 __attribute__((ext_vector_type(8)));

namespace {
constexpr int   kB     = 32;
constexpr int   kA     = 8732;
constexpr int   kC     = 81;
constexpr int   kFG    = 80;     // foreground classes
constexpr int   kTopK  = 100;
constexpr int   kPad   = 112;    // 7 WMMA 16x16 tiles
constexpr float kScoreThr = 0.3f;
constexpr float kNmsThr   = 0.6f;
}

__device__ __forceinline__ unsigned f2key(float f) {
  unsigned u = __float_as_uint(f);
  return (u & 0x80000000u) ? ~u : (u | 0x80000000u);
}

// ---------------------------------------------------------------------------
// Kernel 1: decode boxes (B*A threads, coalesced)
// ---------------------------------------------------------------------------
__global__ __launch_bounds__(256) void decode_kernel(
    const float* __restrict__ reg,      // (B, A, 4)
    const float* __restrict__ priors,   // (A, 4)
    float* __restrict__ boxes)          // (B, A, 4) decoded
{
  int n = blockIdx.x * blockDim.x + threadIdx.x;
  if (n >= kB * kA) return;
  int a = n % kA;
  const float* r = reg + (size_t)n * 4;
  const float* p = priors + (size_t)a * 4;
  float ty = r[0] * 0.1f;          // / 10
  float tx = r[1] * 0.1f;
  float th = r[2] * 0.2f;          // / 5
  float tw = r[3] * 0.2f;
  float cy = ty * p[2] + p[0];
  float cx = tx * p[3] + p[1];
  float h  = __expf(th) * p[2];
  float w  = __expf(tw) * p[3];
  float* o = boxes + (size_t)n * 4;
  o[0] = cy - 0.5f * h;
  o[1] = cx - 0.5f * w;
  o[2] = cy + 0.5f * h;
  o[3] = cx + 0.5f * w;
}

// ---------------------------------------------------------------------------
// Kernel 2: per-(image,class) exact top-100 by radix select in LDS.
// The strided (stride 81*4=324B) score column is gathered into LDS by the
// Tensor Data Mover (TDM) instead of wasting vector-load slots.
// ---------------------------------------------------------------------------
__global__ __launch_bounds__(256) void topk_kernel(
    const float* __restrict__ logits,   // (B, A, C)
    float* __restrict__ tkS,            // (B*FG, 100) masked scores
    int*   __restrict__ tkI)            // (B*FG, 100) anchor indices
{
  __shared__ float    sc[kA];           // 34928 B: raw logits -> masked scores
  __shared__ unsigned hist[256];
  __shared__ float    candS[256];
  __shared__ int      candI[256];
  __shared__ unsigned s_cnt, s_prefix, s_want;

  const int tid = threadIdx.x;
  const int bc  = blockIdx.x;           // 0 .. B*FG-1
  const int img = bc / kFG;
  const int cls = bc % kFG;             // maps to logits channel cls+1

  // ---- TDM: DMA the strided score column (1 x 8732 tile, stride 81) to LDS
  if (tid < 32) {                       // wave 0 only (wave32)
    uint64_t gaddr = (uint64_t)(uintptr_t)logits +
                     (((uint64_t)img * kA) * kC + (uint64_t)(cls + 1)) * 4ull;
    uint32_t laddr = (uint32_t)(uintptr_t)&sc[0];   // flat addr low 32 = LDS offset

    u32x4 g0;
    g0.x = 1u;                                        // count=1, user descriptor
    g0.y = laddr;                                     // lds_addr
    g0.z = (uint32_t)(gaddr & 0xFFFFFFFFu);           // global_addr[31:0]
    g0.w = (uint32_t)((gaddr >> 32) & 0x01FFFFFFu)    // global_addr[56:32]
         | (2u << 30);                                // type = 2 ("image")

    i32x8 g1;
    g1[0] = (int)(2u << 16);              // data_size = 4 bytes
    g1[1] = (int)(1u << 16);              // tensor_dim0 = 1 (low 16)
    g1[2] = (int)((unsigned)kA << 16);    // tensor_dim0 hi=0 | tensor_dim1 lo = 8732
    g1[3] = (int)(1u << 16);              // tensor_dim1 hi=0 | tile_dim0 = 1
    g1[4] = (int)kA;                      // tile_dim1 = 8732 | tile_dim2 = 0
    g1[5] = (int)kC;                      // tensor_dim0_stride = 81
    g1[6] = 0;                            // stride hi | tensor_dim1_stride lo
    g1[7] = 0;

    i32x4 gz4 = {0, 0, 0, 0};
    i32x8 gz8 = {0, 0, 0, 0, 0, 0, 0, 0};
    __builtin_amdgcn_tensor_load_to_lds(g0, g1, gz4, gz4, gz8, 0);
    __builtin_amdgcn_s_wait_tensorcnt(0);
  }
  __syncthreads();

  // ---- sigmoid + threshold mask, in place
  for (int a = tid; a < kA; a += 256) {
    float x = sc[a];
    float p = 1.0f / (1.0f + __expf(-x));
    sc[a] = (p > kScoreThr) ? p : -1.0f;
  }
  if (tid == 0) { s_prefix = 0u; s_want = (unsigned)kTopK; s_cnt = 0u; }
  __syncthreads();

  // ---- 4-pass MSB-first radix select for the exact 100th-largest key
  for (int shift = 24; shift >= 0; shift -= 8) {
    hist[tid] = 0u;
    __syncthreads();
    unsigned pref   = s_prefix;
    unsigned maskHi = (shift == 24) ? 0u : (0xFFFFFFFFu << (shift + 8));
    for (int a = tid; a < kA; a += 256) {
      unsigned k = f2key(sc[a]);
      if ((k & maskHi) == pref)
        atomicAdd(&hist[(k >> shift) & 255u], 1u);
    }
    __syncthreads();
    if (tid == 0) {
      unsigned acc = 0u, want = s_want, bin = 0u;
      for (int b = 255; b >= 0; --b) {
        unsigned h = hist[b];
        if (acc + h >= want) { bin = (unsigned)b; s_want = want - acc; break; }
        acc += h;
      }
      s_prefix = pref | (bin << shift);
    }
    __syncthreads();
  }
  const unsigned pivot = s_prefix;

  // ---- collect: strictly greater (<100), then pivot-equal (tie group)
  for (int a = tid; a < kA; a += 256) {
    unsigned k = f2key(sc[a]);
    if (k > pivot) {
      unsigned p = atomicAdd(&s_cnt, 1u);
      if (p < 256u) { candS[p] = sc[a]; candI[p] = a; }
    }
  }
  __syncthreads();
  for (int a = tid; a < kA; a += 256) {
    unsigned k = f2key(sc[a]);
    if (k == pivot) {
      unsigned p = atomicAdd(&s_cnt, 1u);
      if (p < 256u) { candS[p] = sc[a]; candI[p] = a; }
    }
  }
  __syncthreads();
  unsigned total = s_cnt < 256u ? s_cnt : 256u;
  if ((unsigned)tid >= total) { candS[tid] = -3.0f; candI[tid] = 0x7FFFFFFF; }
  __syncthreads();

  // ---- bitonic sort of 256 (score desc, index asc) — stable-top_k order
  for (unsigned k = 2; k <= 256; k <<= 1) {
    for (unsigned j = k >> 1; j > 0; j >>= 1) {
      __syncthreads();
      unsigned i = (unsigned)tid, ixj = i ^ j;
      if (ixj > i) {
        float si = candS[i], sj = candS[ixj];
        int   ii = candI[i], ij = candI[ixj];
        bool before = (si > sj) || (si == sj && ii < ij);
        bool dirUp  = ((i & k) == 0u);
        if (dirUp ? !before : before) {
          candS[i] = sj; candS[ixj] = si;
          candI[i] = ij; candI[ixj] = ii;
        }
      }
    }
  }
  __syncthreads();

  if (tid < kTopK) {
    float s = candS[tid];
    int idx = candI[tid];
    if (idx < 0 || idx >= kA) idx = 0;   // padding slot (invalid anyway)
    tkS[(size_t)bc * kTopK + tid] = s;
    tkI[(size_t)bc * kTopK + tid] = idx;
  }
}

// ---------------------------------------------------------------------------
// Kernel 3: per-(image,class) NMS. Pairwise area-sum (IoU denominator term)
// computed as a rank-2 outer product on the WMMA pipe:
//   [area_i, 1] x [[1],[area_j]]  ->  area_i + area_j   (16x16x4 f32 tiles)
// ---------------------------------------------------------------------------
__global__ __launch_bounds__(256) void nms_kernel(
    const float* __restrict__ boxes,    // (B, A, 4) decoded
    const float* __restrict__ tkS,
    const int*   __restrict__ tkI,
    float* __restrict__ out)            // (B, FG, 100, 5)
{
  __shared__ float mat[kPad * kPad];    // areaSum, then IoU (50176 B)
  __shared__ float bx[kPad * 4];
  __shared__ float area_s[kPad];
  __shared__ float sc_s[kPad];
  __shared__ int   keep_s[kPad];

  const int tid = threadIdx.x;
  const int bc  = blockIdx.x;
  const int img = bc / kFG;

  // ---- gather top-k boxes into LDS (pad 100 -> 112 with zeros)
  if (tid < kPad) {
    float s = -1.0f, b0 = 0.f, b1 = 0.f, b2 = 0.f, b3 = 0.f;
    if (tid < kTopK) {
      s = tkS[(size_t)bc * kTopK + tid];
      int idx = tkI[(size_t)bc * kTopK + tid];
      const float* bp = boxes + ((size_t)img * kA + (size_t)idx) * 4;
      b0 = bp[0]; b1 = bp[1]; b2 = bp[2]; b3 = bp[3];
    }
    bx[tid * 4 + 0] = b0; bx[tid * 4 + 1] = b1;
    bx[tid * 4 + 2] = b2; bx[tid * 4 + 3] = b3;
    sc_s[tid]  = s;
    float dh = b2 - b0; if (dh < 0.f) dh = 0.f;
    float dw = b3 - b1; if (dw < 0.f) dw = 0.f;
    area_s[tid] = dh * dw;
    keep_s[tid] = 1;
  }
  __syncthreads();

  // ---- WMMA: mat[i][j] = area[i] + area[j], 7x7 tiles of 16x16, K=4
  {
    const int wave = tid >> 5;
    const int lane = tid & 31;
    const int m    = lane & 15;
    const bool lo  = (lane < 16);
    for (int t = wave; t < 49; t += 8) {
      const int ti = t / 7, tj = t % 7;
      // A (16x4): lanes 0-15 hold K=0,1 ; lanes 16-31 hold K=2,3
      v2f afrag;
      afrag.x = lo ? area_s[ti * 16 + m] : 0.0f;  // K0 (or K2=0)
      afrag.y = lo ? 1.0f                : 0.0f;  // K1 (or K3=0)
      // B (4x16): VGPR0 = rows K=0 (lanes 0-15) / K=2 ; VGPR1 = K=1 / K=3
      v2f bfrag;
      bfrag.x = lo ? 1.0f                : 0.0f;
      bfrag.y = lo ? area_s[tj * 16 + m] : 0.0f;
      v8f c = {};
      v8f d = __builtin_amdgcn_wmma_f32_16x16x4_f32(
          false, afrag, false, bfrag, (short)0, c, false, false);
      const int col = tj * 16 + m;
      #pragma unroll
      for (int v = 0; v < 8; ++v) {
        int row = ti * 16 + (lo ? v : v + 8);
        mat[row * kPad + col] = d[v];
      }
    }
  }
  __syncthreads();

  // ---- fold in intersection -> IoU, in place
  for (int e = tid; e < kPad * kPad; e += 256) {
    int i = e / kPad, j = e % kPad;
    float ty0 = fmaxf(bx[i * 4 + 0], bx[j * 4 + 0]);
    float tx0 = fmaxf(bx[i * 4 + 1], bx[j * 4 + 1]);
    float by1 = fminf(bx[i * 4 + 2], bx[j * 4 + 2]);
    float bx1 = fminf(bx[i * 4 + 3], bx[j * 4 + 3]);
    float hh = by1 - ty0; if (hh < 0.f) hh = 0.f;
    float ww = bx1 - tx0; if (ww < 0.f) ww = 0.f;
    float inter = hh * ww;
    float asum  = mat[e];                 // area[i] + area[j] (from WMMA)
    mat[e] = inter / (asum - inter + 1e-9f);
  }
  __syncthreads();

  // ---- sequential greedy suppression (matches fori_loop semantics)
  for (int i = 0; i < kTopK; ++i) {
    bool cur = (keep_s[i] != 0) && (sc_s[i] > kScoreThr);
    if (cur && tid > i && tid < kTopK) {
      if (mat[i * kPad + tid] > kNmsThr) keep_s[tid] = 0;
    }
    __syncthreads();
  }

  // ---- emit (B, FG, 100, 5)
  if (tid < kTopK) {
    bool kf = (keep_s[tid] != 0) && (sc_s[tid] > kScoreThr);
    size_t base = ((size_t)bc * kTopK + tid) * 5;
    out[base + 0] = kf ? bx[tid * 4 + 0] : 0.0f;
    out[base + 1] = kf ? bx[tid * 4 + 1] : 0.0f;
    out[base + 2] = kf ? bx[tid * 4 + 2] : 0.0f;
    out[base + 3] = kf ? bx[tid * 4 + 3] : 0.0f;
    out[base + 4] = kf ? sc_s[tid]       : 0.0f;
  }
}

// ---------------------------------------------------------------------------
extern "C" void kernel_launch(void* const* d_in, const int* in_sizes, int n_in,
                              void* d_out, int out_size, void* d_ws, size_t ws_size,
                              hipStream_t stream) {
  (void)in_sizes; (void)n_in; (void)out_size; (void)ws_size;
  const float* logits = (const float*)d_in[0];   // (B, A, C)
  const float* reg    = (const float*)d_in[1];   // (B, A, 4)
  const float* priors = (const float*)d_in[2];   // (A, 4)
  float* out = (float*)d_out;                    // (B, FG, 100, 5)

  // workspace layout
  float* boxes = (float*)d_ws;                              // B*A*4 floats
  float* tkS   = boxes + (size_t)kB * kA * 4;               // B*FG*100 floats
  int*   tkI   = (int*)(tkS + (size_t)kB * kFG * kTopK);    // B*FG*100 ints

  decode_kernel<<<(kB * kA + 255) / 256, 256, 0, stream>>>(reg, priors, boxes);
  topk_kernel<<<kB * kFG, 256, 0, stream>>>(logits, tkS, tkI);
  nms_kernel<<<kB * kFG, 256, 0, stream>>>(boxes, tkS, tkI, out);
}